// GNN_layer_49400713838636
// MI455X (gfx1250) — compile-verified
//
#include <hip/hip_runtime.h>

typedef __attribute__((ext_vector_type(2))) float v2f;
typedef __attribute__((ext_vector_type(4))) float v4f;
typedef __attribute__((ext_vector_type(8))) float v8f;

#define IN_DIM   256
#define OUT_DIM  128
#define LDS_STRIDE 260   // 256 + 4 pad: row stride % 64 banks == 4 -> conflict-free
#define LEAKY    0.01f

__device__ __forceinline__ float leaky(float x) {
    return (x > 0.f) ? x : LEAKY * x;
}

// ---------------------------------------------------------------------------
// Kernel 1: support = LeakyReLU(features @ W^T) using V_WMMA_F32_16X16X4_F32
// Block: 256 threads (8 wave32). Block tile: 16 rows x 128 cols.
// Wave w computes the 16x16 tile at columns [16w, 16w+16).
// ---------------------------------------------------------------------------
__global__ __launch_bounds__(256) void gemm_lrelu_kernel(
    const float* __restrict__ feat,     // [n_rows, 256] row-major
    const float* __restrict__ W,        // [128, 256] row-major (out, in)
    float* __restrict__ support,        // [n_rows, 128]
    int n_rows)
{
    __shared__ float lds_a[16 * LDS_STRIDE];

    const int tid  = threadIdx.x;
    const int wave = tid >> 5;
    const int lane = tid & 31;
    const int row0 = blockIdx.x * 16;
    const bool full_tile = (row0 + 16) <= n_rows;   // block-uniform

    // Cooperative stage of the 16x256 feature tile into LDS (float4 loads).
    if (full_tile) {
        #pragma unroll
        for (int s = 0; s < 4; ++s) {
            const int i  = tid + s * 256;
            const int r  = i >> 6;        // 0..15
            const int c4 = i & 63;        // float4 column
            const v4f f = ((const v4f*)feat)[(size_t)(row0 + r) * (IN_DIM / 4) + c4];
            float* dst = &lds_a[r * LDS_STRIDE + c4 * 4];
            dst[0] = f.x; dst[1] = f.y; dst[2] = f.z; dst[3] = f.w;
        }
    } else {
        for (int i = tid; i < 16 * (IN_DIM / 4); i += 256) {
            const int r   = i >> 6;
            const int c4  = i & 63;
            const int row = row0 + r;
            v4f f = {0.f, 0.f, 0.f, 0.f};
            if (row < n_rows)
                f = ((const v4f*)feat)[(size_t)row * (IN_DIM / 4) + c4];
            float* dst = &lds_a[r * LDS_STRIDE + c4 * 4];
            dst[0] = f.x; dst[1] = f.y; dst[2] = f.z; dst[3] = f.w;
        }
    }
    __syncthreads();

    // Fragment addressing (ISA 7.12.2, 32-bit A 16x4 / mirrored B 4x16):
    //   lane L holds row/col = L%16, K-pair = 2*(L/16) + {0,1} per VGPR.
    const int n0    = wave * 16;
    const int mrow  = lane & 15;          // M for A, N for B
    const int khalf = (lane >> 4) << 1;   // 0 or 2

    const float* a_base = &lds_a[mrow * LDS_STRIDE + khalf];
    const float* b_base = &W[(size_t)(n0 + mrow) * IN_DIM + khalf];

    v8f acc = {};
    #pragma unroll 8
    for (int kb = 0; kb < IN_DIM / 4; ++kb) {
        const int k0 = kb * 4;
        v2f a; a.x = a_base[k0]; a.y = a_base[k0 + 1];
        v2f b; b.x = b_base[k0]; b.y = b_base[k0 + 1];
        acc = __builtin_amdgcn_wmma_f32_16x16x4_f32(
            /*neg_a=*/false, a, /*neg_b=*/false, b,
            /*c_mod=*/(short)0, acc, /*reuse_a=*/false, /*reuse_b=*/false);
    }

    // C/D layout: VGPR i -> M = i + 8*(lane/16), N = lane%16.
    const int mbase = (lane >> 4) * 8;
    float* sp = &support[(size_t)(row0 + mbase) * OUT_DIM + n0 + mrow];
    if (full_tile) {
        #pragma unroll
        for (int i = 0; i < 8; ++i)
            sp[i * OUT_DIM] = leaky(acc[i]);
    } else {
        #pragma unroll
        for (int i = 0; i < 8; ++i) {
            if (row0 + mbase + i < n_rows)
                sp[i * OUT_DIM] = leaky(acc[i]);
        }
    }
}

// ---------------------------------------------------------------------------
// Kernel 2: CSR row pointers via binary search on sorted edge_rows.
// rp[r] = lower_bound(edge_rows, r);  rp[n_rows] == n_edges.
// ---------------------------------------------------------------------------
__global__ void build_rowptr_kernel(const int* __restrict__ rows,
                                    int* __restrict__ rp,
                                    int n_rows, int n_edges)
{
    const int r = blockIdx.x * blockDim.x + threadIdx.x;
    if (r > n_rows) return;
    int lo = 0, hi = n_edges;
    while (lo < hi) {
        const int mid = (lo + hi) >> 1;
        if (rows[mid] < r) lo = mid + 1; else hi = mid;
    }
    rp[r] = lo;
}

// ---------------------------------------------------------------------------
// Kernel 3: SpMM. One wave32 per output row; lane owns a v4f (4 dims).
// Edge (col, val) pairs loaded 32-at-a-time and broadcast via __shfl.
// Output stored non-temporal (write-once, never re-read on device).
// ---------------------------------------------------------------------------
__global__ __launch_bounds__(256) void spmm_kernel(
    const float* __restrict__ support,  // [n_rows, 128]
    const int* __restrict__ rp,         // [n_rows+1]
    const int* __restrict__ cols,       // [E]
    const float* __restrict__ vals,     // [E]
    float* __restrict__ out,            // [n_rows, 128]
    int n_rows)
{
    const int lane = threadIdx.x & 31;
    const int wave = threadIdx.x >> 5;
    const int row  = blockIdx.x * 8 + wave;
    if (row >= n_rows) return;

    const int start = rp[row];
    const int end   = rp[row + 1];

    const v4f* s4 = (const v4f*)support;
    v4f acc = {0.f, 0.f, 0.f, 0.f};

    for (int base = start; base < end; base += 32) {
        int n = end - base; if (n > 32) n = 32;
        int   c = 0;
        float v = 0.f;
        if (lane < n) {
            c = cols[base + lane];
            v = vals[base + lane];
        }
        for (int j = 0; j < n; ++j) {
            const int   cj = __shfl(c, j);
            const float vj = __shfl(v, j);
            const v4f s = s4[(size_t)cj * (OUT_DIM / 4) + lane];
            acc += vj * s;
        }
    }
    __builtin_nontemporal_store(acc, &((v4f*)out)[(size_t)row * (OUT_DIM / 4) + lane]);
}

// ---------------------------------------------------------------------------
extern "C" void kernel_launch(void* const* d_in, const int* in_sizes, int n_in,
                              void* d_out, int out_size, void* d_ws, size_t ws_size,
                              hipStream_t stream)
{
    const float* feat  = (const float*)d_in[0];   // [N, 256]
    const float* W     = (const float*)d_in[1];   // [128, 256]
    const int*   erows = (const int*)  d_in[2];   // [E] sorted
    const int*   ecols = (const int*)  d_in[3];   // [E]
    const float* evals = (const float*)d_in[4];   // [E]
    float*       out   = (float*)d_out;           // [N, 128]

    const int n_rows  = in_sizes[0] / IN_DIM;
    const int n_edges = in_sizes[2];

    float* support = (float*)d_ws;
    int*   rp      = (int*)((char*)d_ws + (size_t)n_rows * OUT_DIM * sizeof(float));

    // Phase 1: WMMA GEMM + LeakyReLU into workspace
    gemm_lrelu_kernel<<<(n_rows + 15) / 16, 256, 0, stream>>>(feat, W, support, n_rows);

    // Phase 2: row pointers from sorted edge_rows
    build_rowptr_kernel<<<(n_rows + 1 + 255) / 256, 256, 0, stream>>>(erows, rp, n_rows, n_edges);

    // Phase 3: bandwidth-bound SpMM, one wave per row
    spmm_kernel<<<(n_rows + 7) / 8, 256, 0, stream>>>(support, rp, ecols, evals, out, n_rows);
}